// DenseTransformerConv_38259568672922
// MI455X (gfx1250) — compile-verified
//
#include <hip/hip_runtime.h>
#include <hip/hip_bf16.h>
#include <math.h>

// ---------------------------------------------------------------------------
// DenseTransformerConv for MI455X (gfx1250), fp32 WMMA + async-to-LDS path.
// B=4, N=256, D_MODEL=256, D_EDGE=64, H=8, C=32.
// ---------------------------------------------------------------------------

typedef float v2f __attribute__((ext_vector_type(2)));
typedef float v8f __attribute__((ext_vector_type(8)));

#define NN      256
#define DM      256
#define DE      64
#define LDE     68              // padded LDS row stride for edge tile (bank-conflict free)
#define NH      8
#define SCS     9               // padded LDS row stride for scores (gcd(9,64)=1)
#define CC      32
#define ROWS    1024            // B*N
#define SCALE   0.17677669529663687f   // 1/sqrt(32)

__device__ __forceinline__ v8f wmma_f32(v2f a, v2f b, v8f c) {
    // D = A(16x4 f32) * B(4x16 f32) + C(16x16 f32)
    return __builtin_amdgcn_wmma_f32_16x16x4_f32(
        /*neg_a=*/false, a, /*neg_b=*/false, b,
        /*c_mod=*/(short)0, c, /*reuse_a=*/false, /*reuse_b=*/false);
}

// 16B async copy global -> LDS (ASYNCcnt-tracked, no VGPR staging).
__device__ __forceinline__ void async_copy_b128(uint32_t lds_byte_addr, const void* gaddr) {
    asm volatile("global_load_async_to_lds_b128 %0, %1, off"
                 :: "v"(lds_byte_addr), "v"(gaddr) : "memory");
}
__device__ __forceinline__ void wait_async0() {
#if __has_builtin(__builtin_amdgcn_s_wait_asynccnt)
    __builtin_amdgcn_s_wait_asynccnt(0);
#else
    asm volatile("s_wait_asynccnt 0" ::: "memory");
#endif
}

// ---------------------------------------------------------------------------
// Kernel 1: fused projections  Y = x @ [Wq | Wk | Wv | Ws] + bias
//   grid = (64, 8); block = 256 threads = 8 waves; wave owns a 16x16 D tile.
//   Weight tile (256x16) async-staged to LDS once, shared by all 8 waves.
// ---------------------------------------------------------------------------
__global__ __launch_bounds__(256) void dtc_proj_kernel(
    const float* __restrict__ x,
    const float* __restrict__ Wq, const float* __restrict__ bq,
    const float* __restrict__ Wk, const float* __restrict__ bk,
    const float* __restrict__ Wv, const float* __restrict__ bv,
    const float* __restrict__ Ws, const float* __restrict__ bs,
    float* __restrict__ Qo, float* __restrict__ Ko,
    float* __restrict__ Vo, float* __restrict__ So)
{
    __shared__ float sW[DM * 16];     // 16 KB: B tile, rows k=0..255, cols 0..15

    const int lane = threadIdx.x & 31;
    const int wave = threadIdx.x >> 5;
    const int hl   = lane >> 4;       // lane half selects K pair / M half
    const int r    = lane & 15;

    const int mBase = blockIdx.y * 128 + wave * 16;
    const int nb    = blockIdx.x;     // 0..63
    const int mat   = nb >> 4;        // which weight matrix
    const int nLoc  = (nb & 15) * 16 + r;

    const float* W; const float* bias; float* dst;
    if      (mat == 0) { W = Wq; bias = bq; dst = Qo; }
    else if (mat == 1) { W = Wk; bias = bk; dst = Ko; }
    else if (mat == 2) { W = Wv; bias = bv; dst = Vo; }
    else               { W = Ws; bias = bs; dst = So; }

    // --- async-stage the 256x16 weight tile into LDS (1024 x b128)
    {
        const float*   Wt   = W + (nb & 15) * 16;
        const uint32_t ldsW = (uint32_t)(uintptr_t)sW;
        #pragma unroll
        for (int v = 0; v < 4; ++v) {
            const int e    = v * 256 + threadIdx.x;   // 0..1023 float4 chunks
            const int krow = e >> 2, part = e & 3;
            async_copy_b128(ldsW + (uint32_t)((krow * 16 + part * 4) * 4),
                            Wt + (size_t)krow * DM + part * 4);
        }
    }
    wait_async0();
    __syncthreads();

    v8f acc = {};
    const float* xrow = x + (size_t)(mBase + r) * DM;
    #pragma unroll 4
    for (int k = 0; k < DM; k += 4) {
        const int kk = k + 2 * hl;
        v2f a = *(const v2f*)(xrow + kk);            // A[m=r][kk..kk+1]
        v2f b;
        b.x = sW[(kk + 0) * 16 + r];                 // B[kk][n]
        b.y = sW[(kk + 1) * 16 + r];
        acc = wmma_f32(a, b, acc);
    }

    const float bn = bias[nLoc];
    #pragma unroll
    for (int rr = 0; rr < 8; ++rr) {
        const int row = mBase + rr + 8 * hl;         // D layout: M split at lane 16
        dst[(size_t)row * DM + nLoc] = acc[rr] + bn;
    }
}

// ---------------------------------------------------------------------------
// Kernel 2: fused edge-biased attention, one workgroup per (b,i).
//   grid = 1024, block = 256 threads (8 waves).
// Edge row tile (256x64 f32) async-copied to LDS, read from HBM exactly once;
// the fetch overlaps the P = We.Q precompute.
// ---------------------------------------------------------------------------
__global__ __launch_bounds__(256) void dtc_attn_kernel(
    const float* __restrict__ edge,          // (B,N,N,DE)
    const unsigned char* __restrict__ mask,  // (B,N,N) bool
    const float* __restrict__ We,            // (DE, DM)
    const float* __restrict__ Qw, const float* __restrict__ Kw,
    const float* __restrict__ Vw,
    float* __restrict__ out)                 // (B,N,DM), pre-seeded with skip
{
    extern __shared__ float smem[];
    float* sEdge = smem;                 // 256 rows x LDE(68) = 17408 f32
    float* sP    = sEdge + NN * LDE;     // 8*64 = 512 (P, later reused as G)
    float* sQ    = sP + NH * DE;         // 256
    float* sSc   = sQ + DM;              // 256 rows x SCS(9) = 2304 (scores->alpha)

    const int tid  = threadIdx.x;
    const int lane = tid & 31;
    const int wave = tid >> 5;
    const int hl   = lane >> 4;
    const int r    = lane & 15;

    const int bi = blockIdx.x;           // b*N + i
    const int b  = bi >> 8;

    // --- async-stage edge_attr[b,i,:,:] into LDS (4096 x b128, padded rows)
    {
        const float*   gE   = edge + (size_t)bi * NN * DE;
        const uint32_t ldsE = (uint32_t)(uintptr_t)sEdge;
        #pragma unroll
        for (int v = 0; v < 16; ++v) {
            const int e   = v * 256 + tid;           // 0..4095 float4 chunks
            const int row = e >> 4, part = e & 15;   // 16 chunks per 64-float row
            async_copy_b128(ldsE + (uint32_t)((row * LDE + part * 4) * 4),
                            gE + (size_t)e * 4);
        }
    }
    sQ[tid] = Qw[(size_t)bi * DM + tid];
    __syncthreads();                     // sQ visible (edge still in flight)

    // --- P[h][d] = sum_c Q_i[h,c] * We[d][h*32+c]   (overlaps edge async copy)
    for (int t = tid; t < NH * DE; t += 256) {
        const int h = t >> 6, d = t & 63;
        const float* wrow = We + (size_t)d * DM + h * CC;
        const float* q    = sQ + h * CC;
        float s = 0.f;
        #pragma unroll
        for (int c = 0; c < CC; ++c) s = fmaf(q[c], wrow[c], s);
        sP[t] = s;                        // t == h*64 + d
    }
    wait_async0();                        // own edge chunks landed in LDS
    __syncthreads();                      // everyone's chunks + P visible

    // --- scores[j][h] = scale * ( edge_j . P_h  +  K_j[h,:] . Q_i[h,:] )
    // Two WMMA accumulations into one fp32 tile: A rows = j, B cols = h (8 of 16).
    const unsigned char* mrow = mask + (size_t)bi * NN;
    const float* Kbase = Kw + (size_t)b * NN * DM;
    for (int jt = wave; jt < 16; jt += 8) {
        const int mBase = jt * 16;
        v8f acc = {};
        // (a) edge(256x64) @ P^T(64x8)
        for (int k = 0; k < DE; k += 4) {
            const int kk = k + 2 * hl;
            v2f a = *(const v2f*)(sEdge + (mBase + r) * LDE + kk);
            v2f bf;
            bf.x = (r < NH) ? sP[r * DE + kk + 0] : 0.f;
            bf.y = (r < NH) ? sP[r * DE + kk + 1] : 0.f;
            acc = wmma_f32(a, bf, acc);
        }
        // (b) K(256x256) @ blockdiag(Q_i)(256x8): B[k][h] = (k/32==h) ? Q[k] : 0
        for (int k = 0; k < DM; k += 4) {
            const int kk = k + 2 * hl;
            v2f a = *(const v2f*)(Kbase + (size_t)(mBase + r) * DM + kk);
            v2f bf;
            bf.x = (((kk + 0) >> 5) == r) ? sQ[kk + 0] : 0.f;
            bf.y = (((kk + 1) >> 5) == r) ? sQ[kk + 1] : 0.f;
            acc = wmma_f32(a, bf, acc);
        }
        if (r < NH) {
            #pragma unroll
            for (int rr = 0; rr < 8; ++rr) {
                const int j = mBase + rr + 8 * hl;
                const float v = acc[rr] * SCALE;
                sSc[j * SCS + r] = mrow[j] ? v : -INFINITY;
            }
        }
    }
    __syncthreads();

    // --- softmax over j per head; wave w handles head w (wave32 shuffles)
    {
        const int h = wave;
        float loc[8];
        float m = -INFINITY;
        #pragma unroll
        for (int u = 0; u < 8; ++u) {
            loc[u] = sSc[(lane + 32 * u) * SCS + h];
            m = fmaxf(m, loc[u]);
        }
        for (int off = 16; off; off >>= 1) m = fmaxf(m, __shfl_xor(m, off, 32));
        float ssum = 0.f;
        if (m > -INFINITY) {
            #pragma unroll
            for (int u = 0; u < 8; ++u) { loc[u] = __expf(loc[u] - m); ssum += loc[u]; }
        } else {
            #pragma unroll
            for (int u = 0; u < 8; ++u) loc[u] = 0.f;   // fully-masked row -> 0
        }
        for (int off = 16; off; off >>= 1) ssum += __shfl_xor(ssum, off, 32);
        const float inv = (ssum > 0.f) ? 1.f / ssum : 0.f;
        #pragma unroll
        for (int u = 0; u < 8; ++u) sSc[(lane + 32 * u) * SCS + h] = loc[u] * inv;
    }
    __syncthreads();

    // --- G = alpha^T(16x256, rows h<8) @ edge(256x64) via WMMA; waves 0..3
    //     take one 16-wide N tile each. Result overwrites sP (P is consumed).
    if (wave < 4) {
        const int nBase = wave * 16;
        v8f acc = {};
        for (int k = 0; k < NN; k += 4) {
            const int kk = k + 2 * hl;
            v2f a;
            a.x = (r < NH) ? sSc[(kk + 0) * SCS + r] : 0.f;  // A[h=r][j=kk]
            a.y = (r < NH) ? sSc[(kk + 1) * SCS + r] : 0.f;
            v2f bf;
            bf.x = sEdge[(kk + 0) * LDE + nBase + r];        // B[j][d]
            bf.y = sEdge[(kk + 1) * LDE + nBase + r];
            acc = wmma_f32(a, bf, acc);
        }
        if (hl == 0) {                                       // rows 0..7 = heads
            #pragma unroll
            for (int rr = 0; rr < 8; ++rr)
                sP[rr * DE + nBase + r] = acc[rr];
        }
    }
    // --- out_V[h][c] = sum_j alpha[j][h] * V[b,j,h*32+c]  (all threads, VALU)
    const int h = tid >> 5, c = tid & 31;
    const float* Vbase = Vw + (size_t)b * NN * DM + h * CC + c;
    float ov = 0.f;
    for (int j = 0; j < NN; ++j) ov = fmaf(sSc[j * SCS + h], Vbase[(size_t)j * DM], ov);
    __syncthreads();                                         // G complete
    // --- out_E[h][c] = sum_d G[h][d] * We[d][h*32+c]
    float oe = 0.f;
    {
        const float* wcol = We + h * CC + c;
        const float* g    = sP + h * DE;
        #pragma unroll
        for (int d = 0; d < DE; ++d) oe = fmaf(g[d], wcol[(size_t)d * DM], oe);
    }
    out[(size_t)bi * DM + tid] += ov + oe;   // skip connection already in out
}

// ---------------------------------------------------------------------------
extern "C" void kernel_launch(void* const* d_in, const int* in_sizes, int n_in,
                              void* d_out, int out_size, void* d_ws, size_t ws_size,
                              hipStream_t stream) {
    (void)in_sizes; (void)n_in; (void)out_size; (void)ws_size;
    const float*         x    = (const float*)d_in[0];
    const float*         edge = (const float*)d_in[1];
    const unsigned char* mask = (const unsigned char*)d_in[2];  // jax bool = 1 byte
    const float* Wq = (const float*)d_in[3];
    const float* bq = (const float*)d_in[4];
    const float* Wk = (const float*)d_in[5];
    const float* bk = (const float*)d_in[6];
    const float* Wv = (const float*)d_in[7];
    const float* bv = (const float*)d_in[8];
    const float* We = (const float*)d_in[9];
    const float* Ws = (const float*)d_in[10];
    const float* bs = (const float*)d_in[11];
    float* out = (float*)d_out;

    float* Qw = (float*)d_ws;                 // 1024*256 f32 = 1 MB
    float* Kw = Qw + (size_t)ROWS * DM;       // 1 MB
    float* Vw = Kw + (size_t)ROWS * DM;       // 1 MB

    // Kernel 1: projections (Q,K,V -> workspace; skip -> d_out)
    dtc_proj_kernel<<<dim3(64, 8), dim3(256), 0, stream>>>(
        x, Wq, bq, Wk, bk, Wv, bv, Ws, bs, Qw, Kw, Vw, out);

    // Kernel 2: fused attention, one WG per (b,i)
    const size_t lds_bytes =
        (size_t)(NN * LDE + NH * DE + DM + NN * SCS) * sizeof(float); // 81920 B
    dtc_attn_kernel<<<dim3(ROWS), dim3(256), lds_bytes, stream>>>(
        edge, mask, We, Qw, Kw, Vw, out);
}